// ContextualAffineMaximizerAuction_28063316312729
// MI455X (gfx1250) — compile-verified
//
#include <hip/hip_runtime.h>

// ContextualAffineMaximizerAuction — MI455X (gfx1250, wave32)
// B=4096, MENU=256 (+1 null), N=8 agents, M=16 items.
// One workgroup (256 threads = 8 waves) per batch element.
// allocs[b] (128 KB) is DMA'd global->LDS by the Tensor Data Mover
// (TENSOR_LOAD_TO_LDS, TENSORcnt), util GEMM runs on V_WMMA_F32_16X16X4_F32.

#define BATCH 4096
#define MENU  256
#define S_TOT 257
#define NAG   8
#define MIT   16
#define NM    128                  // NAG*MIT
#define ALLOC_PER_B (MENU * NM)    // 32768 floats = 128 KB

// LDS budget (floats)
#define LDS_FLOATS (ALLOC_PER_B + S_TOT*NAG + S_TOT + S_TOT + S_TOT + NM + NAG + NAG + 8 + 256 + 4)

typedef __attribute__((ext_vector_type(2))) float v2f;
typedef __attribute__((ext_vector_type(8))) float v8f;
typedef unsigned int u32x4 __attribute__((ext_vector_type(4)));
typedef int          i32x4 __attribute__((ext_vector_type(4)));
typedef int          i32x8 __attribute__((ext_vector_type(8)));

__device__ __forceinline__ float waveSum(float v) {
#pragma unroll
    for (int m = 16; m > 0; m >>= 1) v += __shfl_xor(v, m, 32);
    return v;
}
__device__ __forceinline__ float waveMax(float v) {
#pragma unroll
    for (int m = 16; m > 0; m >>= 1) v = fmaxf(v, __shfl_xor(v, m, 32));
    return v;
}

__global__ __launch_bounds__(256)
void auction_kernel(const float* __restrict__ gBids,    // B,N,M
                    const float* __restrict__ gAllocs,  // B,MENU,N,M
                    const float* __restrict__ gW,       // B,N
                    const float* __restrict__ gBoost,   // B,MENU
                    const float* __restrict__ gTemp,    // scalar
                    float* __restrict__ outChoice,      // B,S_TOT
                    float* __restrict__ outItem,        // B,N,M
                    float* __restrict__ outPay,         // N,B
                    float* __restrict__ outAllocs)      // B,S_TOT,N,M
{
    const int b    = blockIdx.x;
    const int tid  = threadIdx.x;
    const int lane = tid & 31;
    const int wid  = tid >> 5;
    const float temp = *gTemp;

    extern __shared__ __align__(16) float smem[];
    float* sAllocs = smem;                    // 32768  allocs[b], [s][n][m]
    float* sPaw    = sAllocs + ALLOC_PER_B;   // 257*8  per-agent welfare (w-scaled)
    float* sTW     = sPaw + S_TOT * NAG;      // 257    total welfare
    float* sChoice = sTW + S_TOT;             // 257    softmax choice
    float* sBoost  = sChoice + S_TOT;         // 257    boosts (null = 0)
    float* sBids   = sBoost + S_TOT;          // 128
    float* sW      = sBids + NM;              // 8
    float* sCW     = sW + NAG;                // 8      chosen welfare per agent
    float* sRedA   = sCW + NAG;               // 8      per-wave reduction scratch
    float* sRedB   = sRedA + 8;               // 256    block reduction scratch
    float* sScal   = sRedB + 256;             // scalars (alloc_b)

    // ---- Phase 0: small per-b inputs into LDS --------------------------------
    if (tid < NM)  sBids[tid] = gBids[(size_t)b * NM + tid];
    if (tid < NAG) sW[tid]    = gW[(size_t)b * NAG + tid];
    sBoost[tid] = gBoost[(size_t)b * MENU + tid];
    if (tid == 0) sBoost[MENU] = 0.0f;

    // ---- Phase 1: allocs[b] -> LDS --------------------------------------------
#if __has_builtin(__builtin_amdgcn_tensor_load_to_lds)
    // Tensor Data Mover DMA: 1 descriptor, contiguous 32768 x 4B tile.
    if (wid == 0) {
        const unsigned long long gaddr =
            (unsigned long long)(uintptr_t)(gAllocs + (size_t)b * ALLOC_PER_B);
        const unsigned int ldsa = (unsigned int)(uintptr_t)sAllocs; // LDS byte offset

        u32x4 g0;
        g0[0] = 1u;                                        // count=1, no gather
        g0[1] = ldsa;                                      // lds_addr
        g0[2] = (unsigned int)(gaddr & 0xFFFFFFFFu);       // global_addr[31:0]
        g0[3] = (unsigned int)((gaddr >> 32) & 0x01FFFFFFu)
              | (2u << 30);                                // global_addr[56:32] | type=2

        i32x8 g1;
        g1[0] = 0x00020000;                  // workgroup_mask=0, data_size=2 (4B)
        g1[1] = (int)((ALLOC_PER_B & 0xFFFF) << 16);       // tensor_dim0 lo16
        g1[2] = (int)((ALLOC_PER_B >> 16) | (1u << 16));   // tensor_dim0 hi16 | tensor_dim1=1
        g1[3] = (int)((ALLOC_PER_B & 0xFFFF) << 16);       // tile_dim0 = 32768
        g1[4] = 1;                                         // tile_dim1=1, tile_dim2=0
        g1[5] = ALLOC_PER_B;                               // tensor_dim0_stride lo32
        g1[6] = 0;
        g1[7] = 0;

        i32x4 gz4; gz4[0] = 0; gz4[1] = 0; gz4[2] = 0; gz4[3] = 0;
        i32x8 gz8; gz8[0] = 0; gz8[1] = 0; gz8[2] = 0; gz8[3] = 0;
                   gz8[4] = 0; gz8[5] = 0; gz8[6] = 0; gz8[7] = 0;
        // 6-arg toolchain variant: (g0, g1, g2, g3, g4, cpol)
        __builtin_amdgcn_tensor_load_to_lds(g0, g1, gz4, gz4, gz8, 0);
    }
    __builtin_amdgcn_s_wait_tensorcnt(0);    // nop for waves 1..7
    __syncthreads();
    // padded copy-out served from LDS
    {
        float4* oA  = (float4*)(outAllocs + (size_t)b * (S_TOT * NM));
        const float4* sA4 = (const float4*)sAllocs;
#pragma unroll 4
        for (int i = tid; i < ALLOC_PER_B / 4; i += 256)
            oA[i] = sA4[i];
        if (tid < 32) {   // null menu entry: 128 zero floats
            float4 z = make_float4(0.f, 0.f, 0.f, 0.f);
            oA[ALLOC_PER_B / 4 + tid] = z;
        }
    }
#else
    // Fallback: fused global->LDS + copy-out through VGPRs.
    {
        const float4* gA = (const float4*)(gAllocs + (size_t)b * ALLOC_PER_B);
        float4* oA  = (float4*)(outAllocs + (size_t)b * (S_TOT * NM));
        float4* sA4 = (float4*)sAllocs;
#pragma unroll 4
        for (int i = tid; i < ALLOC_PER_B / 4; i += 256) {
            float4 v = gA[i];
            sA4[i] = v;
            oA[i]  = v;
        }
        if (tid < 32) {
            float4 z = make_float4(0.f, 0.f, 0.f, 0.f);
            oA[ALLOC_PER_B / 4 + tid] = z;
        }
    }
    __syncthreads();
#endif

    // ---- Phase 2: util GEMM on fp32 WMMA -------------------------------------
    // util[s,n] = sum_k allocs[s,k] * Bdiag[k,n],  k = n*16+m, K=128
    // Bdiag[k,col] = bids[k] if col == k>>4 else 0  (cols 8..15 are zero pad)
    // Each wave owns s-tiles t0=wid, t1=wid+8; 32 chunks of K=4.
    {
        const int col = lane & 15;           // C column (agent n) / A row (s_local)
        const int hi  = lane >> 4;           // K sub-pair select per ISA layout
        v8f acc0 = {}; v8f acc1 = {};
        const int t0 = wid, t1 = wid + 8;
        const float* aRow0 = sAllocs + (t0 * 16 + col) * NM;
        const float* aRow1 = sAllocs + (t1 * 16 + col) * NM;
#pragma unroll
        for (int c = 0; c < 32; ++c) {
            const int kb = c * 4 + hi * 2;   // even -> float2-aligned LDS loads
            const int n0 = kb >> 4;          // same for kb and kb+1 (kb even)
            // unconditional ds_load_b64 + v_cndmask selects (no exec branching)
            float2 t  = *(const float2*)(sBids + kb);
            v2f bf;
            bf.x = (col == n0) ? t.x : 0.0f;
            bf.y = (col == n0) ? t.y : 0.0f;
            float2 a0 = *(const float2*)(aRow0 + kb);
            float2 a1 = *(const float2*)(aRow1 + kb);
            v2f af0; af0.x = a0.x; af0.y = a0.y;
            v2f af1; af1.x = a1.x; af1.y = a1.y;
            acc0 = __builtin_amdgcn_wmma_f32_16x16x4_f32(false, af0, false, bf,
                                                         (short)0, acc0, false, false);
            acc1 = __builtin_amdgcn_wmma_f32_16x16x4_f32(false, af1, false, bf,
                                                         (short)0, acc1, false, false);
        }
        // C/D layout: VGPR v, lanes 0-15 -> row v, lanes 16-31 -> row v+8
        if (col < NAG) {
            const float wn = sW[col];
#pragma unroll
            for (int v = 0; v < 8; ++v) {
                const int sl = v + 8 * hi;
                sPaw[(t0 * 16 + sl) * NAG + col] = wn * acc0[v];
                sPaw[(t1 * 16 + sl) * NAG + col] = wn * acc1[v];
            }
        }
    }
    if (tid < NAG) sPaw[MENU * NAG + tid] = 0.0f;   // null entry welfare
    __syncthreads();

    // ---- Phase 3: main softmax over 257 menu entries -------------------------
    float tw = 0.0f;
#pragma unroll
    for (int n = 0; n < NAG; ++n) tw += sPaw[tid * NAG + n];
    sTW[tid] = tw;
    if (tid == 0) sTW[MENU] = 0.0f;
    const float logit = (tw + sBoost[tid]) * temp;

    float wm = waveMax(logit);
    if (lane == 0) sRedA[wid] = wm;
    __syncthreads();
    float gmax = 0.0f;                        // null entry logit == 0
#pragma unroll
    for (int k = 0; k < 8; ++k) gmax = fmaxf(gmax, sRedA[k]);
    const float e = __expf(logit - gmax);
    float ws = waveSum(e);
    __syncthreads();                          // done reading sRedA (max)
    if (lane == 0) sRedA[wid] = ws;
    __syncthreads();
    float tot = __expf(-gmax);                // null entry
#pragma unroll
    for (int k = 0; k < 8; ++k) tot += sRedA[k];
    const float inv = 1.0f / tot;
    const float choice = e * inv;
    sChoice[tid] = choice;
    outChoice[(size_t)b * S_TOT + tid] = choice;
    if (tid == 0) {
        const float cn = __expf(-gmax) * inv;
        sChoice[MENU] = cn;
        outChoice[(size_t)b * S_TOT + MENU] = cn;
    }
    __syncthreads();

    // ---- Phase 4: item_allocation (choice-weighted menu mix, LDS only) -------
    if (tid < NM) {
        float acc = 0.0f;
        for (int s = 0; s < MENU; ++s)        // null entry contributes 0
            acc += sChoice[s] * sAllocs[s * NM + tid];
        outItem[(size_t)b * NM + tid] = acc;
    }

    // ---- Phase 5: chosen welfare per agent + alloc_b --------------------------
    {
        const int n = tid & 7;
        const int g = tid >> 3;               // 0..31
        float acc = 0.0f;
        for (int s = g; s < MENU; s += 32)
            acc += sPaw[s * NAG + n] * sChoice[s];
        sRedB[tid] = acc;                     // tid == g*8+n
    }
    float pb  = sChoice[tid] * sBoost[tid];   // alloc_b partial (null adds 0)
    float wsb = waveSum(pb);
    __syncthreads();                          // done reading sRedA (sum)
    if (lane == 0) sRedA[wid] = wsb;
    __syncthreads();
    if (tid < NAG) {
        float cw = 0.0f;
#pragma unroll
        for (int g = 0; g < 32; ++g) cw += sRedB[g * 8 + tid];
        sCW[tid] = cw;
    }
    if (tid == 0) {
        float ab = 0.0f;
#pragma unroll
        for (int k = 0; k < 8; ++k) ab += sRedA[k];
        sScal[0] = ab;
    }
    __syncthreads();

    // ---- Phase 6: leave-one-out payments — one wave per removed agent i ------
    {
        const int i = wid;                    // removed agent
        float cwTot = 0.0f;
#pragma unroll
        for (int n = 0; n < NAG; ++n) cwTot += sCW[n];
        const float allocB = sScal[0];

        float lg[8];
        float lmax = 0.0f;                    // null entry logit == 0
#pragma unroll
        for (int j = 0; j < 8; ++j) {
            const int s = lane + 32 * j;
            const float v = (sTW[s] - sPaw[s * NAG + i] + sBoost[s]) * temp;
            lg[j] = v;
            lmax = fmaxf(lmax, v);
        }
        lmax = waveMax(lmax);

        float se = 0.0f, s1 = 0.0f, s2 = 0.0f;
#pragma unroll
        for (int j = 0; j < 8; ++j) {
            const int s  = lane + 32 * j;
            const float ee = __expf(lg[j] - lmax);
            se += ee;
            s1 += ee * (sTW[s] - sPaw[s * NAG + i]);  // removed welfare at s
            s2 += ee * sBoost[s];                      // removed boost
        }
        se = waveSum(se); s1 = waveSum(s1); s2 = waveSum(s2);
        se += __expf(-lmax);                  // null entry (welfare 0, boost 0)
        if (lane == 0) {
            const float pay = ((s1 + s2) / se - (cwTot - sCW[i]) - allocB) / sW[i];
            outPay[(size_t)i * BATCH + b] = pay;
        }
    }
}

extern "C" void kernel_launch(void* const* d_in, const int* in_sizes, int n_in,
                              void* d_out, int out_size, void* d_ws, size_t ws_size,
                              hipStream_t stream) {
    const float* bids   = (const float*)d_in[0];
    const float* allocs = (const float*)d_in[1];
    const float* w      = (const float*)d_in[2];
    const float* boost  = (const float*)d_in[3];
    const float* temp   = (const float*)d_in[4];

    float* out       = (float*)d_out;
    float* outChoice = out;                                       // B*257
    float* outItem   = outChoice + (size_t)BATCH * S_TOT;         // B*128
    float* outPay    = outItem   + (size_t)BATCH * NM;            // 8*B
    float* outAllocs = outPay    + (size_t)NAG * BATCH;           // B*257*128

    const size_t ldsBytes = (size_t)LDS_FLOATS * sizeof(float);   // ~144 KB
    auction_kernel<<<dim3(BATCH), dim3(256), ldsBytes, stream>>>(
        bids, allocs, w, boost, temp, outChoice, outItem, outPay, outAllocs);
}